// ConvMLayer_86517821213086
// MI455X (gfx1250) — compile-verified
//
#include <hip/hip_runtime.h>
#include <hip/hip_bf16.h>

// ConvMLayer: covariance-metric few-shot scoring head, fp32 end-to-end.
// Phase A: cov[t][way] (64x64) from support via V_WMMA_F32_16X16X4_F32 (G = S^T S).
// Phase B: sim = diag(Qc * cov * Qc^T) via WMMA, fused LeakyReLU + Conv1d(k=441).

typedef float v2f __attribute__((ext_vector_type(2)));
typedef float v8f __attribute__((ext_vector_type(8)));

#define T_DIM 8
#define WQ 75
#define CDIM 64
#define HW 441
#define WAY 5
#define SHOT 5

#define SA 452   // phase-A LDS row stride (SA%64==4 -> conflict-free 16-row fragment loads)
#define SB 456   // phase-B LDS row stride (2*SB%64==16 -> halves hit disjoint banks)
#define CS 68    // cov LDS row stride (CS%64==4)

// ---------------------------------------------------------------------------
// Phase A: per (t,way) covariance. Grid = 40 blocks, 256 threads (8 waves).
// G = S^T S accumulated over K=5*441 (padded to 444/shot); cov=(G-n*mu*mu^T)/440.
// ---------------------------------------------------------------------------
__global__ void cov_kernel(const float* __restrict__ supp, float* __restrict__ covws) {
    extern __shared__ float sm[];
    float* chunk  = sm;              // 64 x SA  (S^T slab for one shot, zero-padded to 444)
    float* colsum = sm + CDIM * SA;  // 64

    const int tid  = threadIdx.x;
    const int lane = tid & 31;
    const int wave = tid >> 5;
    const int l15  = lane & 15;
    const int half = lane >> 4;       // K-pair select per ISA A/B layout
    const int t    = blockIdx.x / WAY;
    const int way  = blockIdx.x % WAY;

    if (tid < CDIM) colsum[tid] = 0.f;

    // 16 output tiles (4x4 of 16x16) over 8 waves -> 2 tiles/wave, accumulators live in VGPRs
    const int tile0 = wave * 2, tile1 = wave * 2 + 1;
    const int mt0 = tile0 >> 2, nt0 = tile0 & 3;
    const int mt1 = tile1 >> 2, nt1 = tile1 & 3;
    v8f acc0 = {}; v8f acc1 = {};

    for (int shot = 0; shot < SHOT; ++shot) {
        __syncthreads();  // WAR: previous iteration's reads of chunk complete
        const float* base = supp + (size_t)((t * (WAY * SHOT) + way * SHOT + shot) * CDIM) * HW;
        for (int idx = tid; idx < CDIM * 444; idx += 256) {
            int c = idx / 444, i = idx - c * 444;
            chunk[c * SA + i] = (i < HW) ? base[c * HW + i] : 0.f;
        }
        __syncthreads();

        // per-channel column sums (for mean term)
        {
            int c = tid & 63, seg = tid >> 6;
            float s = 0.f;
            for (int i = seg * 111; i < seg * 111 + 111; ++i) s += chunk[c * SA + i];
            atomicAdd(&colsum[c], s);
        }

        // G partial: chained fp32 WMMA over this shot's K range (444 = 111 steps of K=4)
        for (int kb = 0; kb < 444; kb += 4) {
            const int k0 = kb + 2 * half;
            v2f a0 = *(const v2f*)&chunk[(mt0 * 16 + l15) * SA + k0];
            v2f b0 = *(const v2f*)&chunk[(nt0 * 16 + l15) * SA + k0];  // B via S^T (same slab)
            acc0 = __builtin_amdgcn_wmma_f32_16x16x4_f32(false, a0, false, b0,
                                                         (short)0, acc0, false, false);
            v2f a1 = *(const v2f*)&chunk[(mt1 * 16 + l15) * SA + k0];
            v2f b1 = *(const v2f*)&chunk[(nt1 * 16 + l15) * SA + k0];
            acc1 = __builtin_amdgcn_wmma_f32_16x16x4_f32(false, a1, false, b1,
                                                         (short)0, acc1, false, false);
        }
    }
    __syncthreads();

    const float inv_n = 1.f / (float)(SHOT * HW);  // 1/2205
    const float inv_d = 1.f / (float)(HW - 1);     // 1/440
    float* dst = covws + (size_t)(t * WAY + way) * CDIM * CDIM;
    // C/D layout: VGPR r, lanes 0-15 -> (M=r, N=lane); lanes 16-31 -> (M=r+8, N=lane-16)
    #pragma unroll
    for (int r = 0; r < 8; ++r) {
        {
            int M = mt0 * 16 + r + 8 * half, N = nt0 * 16 + l15;
            dst[M * CDIM + N] = (acc0[r] - colsum[M] * colsum[N] * inv_n) * inv_d;
        }
        {
            int M = mt1 * 16 + r + 8 * half, N = nt1 * 16 + l15;
            dst[M * CDIM + N] = (acc1[r] - colsum[M] * colsum[N] * inv_n) * inv_d;
        }
    }
}

// ---------------------------------------------------------------------------
// Phase B: per (t,q) block. Grid = 600 blocks, 256 threads (8 waves).
// sim = rowsum((Qc @ cov) .* Qc); score[way] = sum_i lrelu(sim_i)*conv_w[i] + b.
// ---------------------------------------------------------------------------
__global__ void sim_kernel(const float* __restrict__ qf, const float* __restrict__ covws,
                           const float* __restrict__ convw, const float* __restrict__ convb,
                           float* __restrict__ out) {
    extern __shared__ float sm[];
    float* qs    = sm;                       // 64 x SB (Q^T, centered in place; rows padded to 448)
    float* covs  = qs + CDIM * SB;           // 5 x 64 x CS
    float* cwl   = covs + WAY * CDIM * CS;   // 448 (conv_w, zero-padded)
    float* meanv = cwl + 448;                // 64
    float* score = meanv + CDIM;             // 8

    const int tid  = threadIdx.x;
    const int lane = tid & 31;
    const int wave = tid >> 5;
    const int l15  = lane & 15;
    const int half = lane >> 4;
    const int b    = blockIdx.x;             // t*75 + q

    for (int i = tid; i < 448; i += 256) cwl[i] = (i < HW) ? convw[i] : 0.f;
    if (tid < CDIM) meanv[tid] = 0.f;
    if (tid < 8)    score[tid] = 0.f;

    // stage all 5 way-covariances (symmetric; stored row-major, read either way)
    {
        const int tt = b / WQ;
        const float* csrc = covws + (size_t)tt * WAY * CDIM * CDIM;
        for (int idx = tid; idx < WAY * CDIM * CDIM; idx += 256) {
            int w = idx >> 12, rem = idx & 4095, c = rem >> 6, d = rem & 63;
            covs[(w * CDIM + c) * CS + d] = csrc[idx];
        }
    }
    // stage Q^T, zero-pad rows 441..447 (28 M-tiles of 16)
    const float* qbase = qf + (size_t)b * CDIM * HW;
    for (int idx = tid; idx < CDIM * 448; idx += 256) {
        int c = idx / 448, i = idx - c * 448;
        qs[c * SB + i] = (i < HW) ? qbase[c * HW + i] : 0.f;
    }
    __syncthreads();

    // per-channel mean over hw, then center in place
    {
        int c = tid & 63, seg = tid >> 6;
        float s = 0.f;
        for (int i = seg; i < HW; i += 4) s += qs[c * SB + i];
        atomicAdd(&meanv[c], s);
    }
    __syncthreads();
    if (tid < CDIM) meanv[tid] *= (1.f / (float)HW);
    __syncthreads();
    for (int idx = tid; idx < CDIM * HW; idx += 256) {
        int c = idx / HW, i = idx - c * HW;
        qs[c * SB + i] -= meanv[c];
    }
    __syncthreads();

    // 140 units = 5 ways x 28 M-tiles, strided over 8 waves (wave-uniform, EXEC all-ones)
    for (int unit = wave; unit < WAY * 28; unit += 8) {
        const int way = unit / 28, mt = unit % 28;
        const float* covw = covs + way * CDIM * CS;
        float rowacc[8] = {0, 0, 0, 0, 0, 0, 0, 0};

        for (int nt = 0; nt < 4; ++nt) {
            v8f acc = {};
            #pragma unroll
            for (int kb = 0; kb < CDIM; kb += 4) {
                const int k0 = kb + 2 * half;
                v2f a;  // A(m=row_i, k=channel): gather from Q^T (stride SB apart)
                a.x = qs[k0 * SB + mt * 16 + l15];
                a.y = qs[(k0 + 1) * SB + mt * 16 + l15];
                // B(k=c, n=d) = cov[c][d] = cov[d][c] (symmetry -> contiguous float2)
                v2f bb = *(const v2f*)&covw[(nt * 16 + l15) * CS + k0];
                acc = __builtin_amdgcn_wmma_f32_16x16x4_f32(false, a, false, bb,
                                                            (short)0, acc, false, false);
            }
            const int colc = nt * 16 + l15;
            #pragma unroll
            for (int r = 0; r < 8; ++r) {
                const int rowi = mt * 16 + r + 8 * half;
                rowacc[r] += acc[r] * qs[colc * SB + rowi];  // P .* Qc
            }
        }

        // reduce each row across its 16 lanes, fuse LeakyReLU + conv_w dot
        float contrib = 0.f;
        #pragma unroll
        for (int r = 0; r < 8; ++r) {
            float v = rowacc[r];
            v += __shfl_xor(v, 1, 16);
            v += __shfl_xor(v, 2, 16);
            v += __shfl_xor(v, 4, 16);
            v += __shfl_xor(v, 8, 16);
            if (l15 == 0) {
                const int row = mt * 16 + r + 8 * half;
                const float s = (v > 0.f) ? v : 0.2f * v;  // LeakyReLU(0.2)
                contrib += s * cwl[row];                   // cwl[row>=441]==0
            }
        }
        if (l15 == 0) atomicAdd(&score[way], contrib);
    }
    __syncthreads();
    if (tid < WAY) out[b * WAY + tid] = score[tid] + convb[0];
}

// ---------------------------------------------------------------------------
extern "C" void kernel_launch(void* const* d_in, const int* in_sizes, int n_in,
                              void* d_out, int out_size, void* d_ws, size_t ws_size,
                              hipStream_t stream) {
    const float* query   = (const float*)d_in[0];  // (8,75,64,21,21)
    const float* support = (const float*)d_in[1];  // (8,25,64,21,21)
    const float* conv_w  = (const float*)d_in[2];  // (441,)
    const float* conv_b  = (const float*)d_in[3];  // (1,)
    float* out    = (float*)d_out;                 // (8,75,5)
    float* cov_ws = (float*)d_ws;                  // needs 40*64*64*4 = 640 KB scratch

    const size_t ldsA = (size_t)(CDIM * SA + CDIM) * sizeof(float);                 // ~116 KB
    const size_t ldsB = (size_t)(CDIM * SB + WAY * CDIM * CS + 448 + CDIM + 8)
                        * sizeof(float);                                            // ~206 KB

    cov_kernel<<<T_DIM * WAY, 256, ldsA, stream>>>(support, cov_ws);
    sim_kernel<<<T_DIM * WQ, 256, ldsB, stream>>>(query, cov_ws, conv_w, conv_b, out);
}